// GCN_15857019256946
// MI455X (gfx1250) — compile-verified
//
#include <hip/hip_runtime.h>
#include <math.h>

typedef __attribute__((ext_vector_type(2))) float v2f;
typedef __attribute__((ext_vector_type(8))) float v8f;

#define NNODES 100000
#define NEDGES 1600000
#define NOBS 48
#define HID 48
#define NACT 32

// ---------------- degree / dinv ----------------
__global__ void deg_kernel(const int* __restrict__ col, unsigned int* __restrict__ degcnt, int E) {
    int e = blockIdx.x * blockDim.x + threadIdx.x;
    if (e < E) atomicAdd(&degcnt[col[e]], 1u);
}

__global__ void dinv_kernel(const unsigned int* __restrict__ degcnt, float* __restrict__ dinv, int n) {
    int i = blockIdx.x * blockDim.x + threadIdx.x;
    if (i < n) dinv[i] = rsqrtf((float)(degcnt[i] + 1u));   // +1 = self loop, deg >= 1 always
}

// ---------------- WMMA GEMM: out[n][j] = dinv[n] * sum_k A(n,k) * W[k][j] ----------------
// A(n,k) = A[k*SK + n*SN]   (GEMM1: SK=NNODES, SN=1 since x is [K][N]; GEMM2: SK=1, SN=HID)
// One wave computes one 16(node) x 16(j) tile. blockDim.x = 32 * (JTOT/16).
template<int K, int JTOT, long SK, long SN>
__global__ void gemm_wmma(const float* __restrict__ A, const float* __restrict__ W,
                          const float* __restrict__ dinv, float* __restrict__ out) {
    const int lane  = threadIdx.x & 31;
    const int jbase = (threadIdx.x >> 5) * 16;   // wave id -> j tile
    const int nbase = blockIdx.x * 16;
    const int half  = lane >> 4;                 // 0: K=0,1 ; 1: K=2,3
    const int l16   = lane & 15;
    const int arow  = nbase + l16;               // node (M) index for A operand

    v8f acc = {};
#pragma unroll
    for (int kb = 0; kb < K; kb += 4) {
        const int k0 = kb + 2 * half;
        v2f a, b;
        a.x = A[(long)k0       * SK + (long)arow * SN];
        a.y = A[(long)(k0 + 1) * SK + (long)arow * SN];
        b.x = W[k0       * JTOT + jbase + l16];
        b.y = W[(k0 + 1) * JTOT + jbase + l16];
        acc = __builtin_amdgcn_wmma_f32_16x16x4_f32(
            /*neg_a=*/false, a, /*neg_b=*/false, b,
            /*c_mod=*/(short)0, acc, /*reuse_a=*/false, /*reuse_b=*/false);
    }

    // C/D layout: VGPR r -> M = r + 8*(lane>=16), N = lane%16
    const int mbase = half * 8;
#pragma unroll
    for (int r = 0; r < 8; ++r) {
        const int node = nbase + mbase + r;
        out[(long)node * JTOT + jbase + l16] = dinv[node] * acc[r];
    }
}

// ---------------- edge scatter: acc[col[e]][d] += g[row[e]][d] ----------------
template<int D>
__global__ void scatter_kernel(const float* __restrict__ g, const int* __restrict__ row,
                               const int* __restrict__ col, float* __restrict__ acc, long total) {
    long tid = (long)blockIdx.x * blockDim.x + threadIdx.x;
    if (tid >= total) return;
    int e = (int)(tid / D);
    int d = (int)(tid % D);
    atomicAdd(&acc[(long)col[e] * D + d], g[(long)row[e] * D + d]);
}

// ---------------- post conv1: h1 = relu(dinv*(acc1 + g1) + b1), written in-place into acc1 ----------------
__global__ void post1_kernel(float* __restrict__ acc1, const float* __restrict__ g1,
                             const float* __restrict__ dinv, const float* __restrict__ b1, long total) {
    long tid = (long)blockIdx.x * blockDim.x + threadIdx.x;
    if (tid >= total) return;
    int node = (int)(tid / HID);
    int d    = (int)(tid % HID);
    float v = dinv[node] * (acc1[tid] + g1[tid]) + b1[d];
    acc1[tid] = fmaxf(v, 0.0f);
}

// ---------------- final: out[d] = sum_n onehot[n] * (dinv[n]*(acc2+g2) + b2[d]) ----------------
__global__ void final_reduce(const float* __restrict__ g2, const float* __restrict__ acc2,
                             const float* __restrict__ dinv, const float* __restrict__ b2,
                             const float* __restrict__ onehot, float* __restrict__ out, int n) {
    __shared__ float sdata[256];
    const int d   = threadIdx.x & 31;   // action dim
    const int sub = threadIdx.x >> 5;   // node sub-lane 0..7
    float s = 0.0f;
    for (int node = blockIdx.x * 8 + sub; node < n; node += gridDim.x * 8) {
        float w  = onehot[node];
        float di = dinv[node];
        long  base = (long)node * NACT + d;
        s += w * (di * (acc2[base] + g2[base]) + b2[d]);
    }
    sdata[threadIdx.x] = s;
    __syncthreads();
    if (sub == 0) {
        float t = 0.0f;
#pragma unroll
        for (int i = 0; i < 8; ++i) t += sdata[i * 32 + d];
        atomicAdd(&out[d], t);
    }
}

extern "C" void kernel_launch(void* const* d_in, const int* in_sizes, int n_in,
                              void* d_out, int out_size, void* d_ws, size_t ws_size,
                              hipStream_t stream) {
    const float* x      = (const float*)d_in[0];   // [NOBS][NNODES]
    const float* onehot = (const float*)d_in[1];   // [NNODES]
    const float* W1     = (const float*)d_in[2];   // [NOBS][HID]
    const float* b1     = (const float*)d_in[3];   // [HID]
    const float* W2     = (const float*)d_in[4];   // [HID][NACT]
    const float* b2     = (const float*)d_in[5];   // [NACT]
    const int*   eidx   = (const int*)d_in[6];     // [2][NEDGES]
    const int*   row    = eidx;                    // source nodes
    const int*   col    = eidx + NEDGES;           // target nodes
    float* out = (float*)d_out;

    // workspace layout (all fp32-sized slots): acc1 | acc2 | degcnt | dinv | g1 | g2
    float*        acc1   = (float*)d_ws;                    // N*48
    float*        acc2   = acc1 + (long)NNODES * HID;       // N*32
    unsigned int* degcnt = (unsigned int*)(acc2 + (long)NNODES * NACT); // N
    float*        dinv   = (float*)(degcnt + NNODES);       // N
    float*        g1     = dinv + NNODES;                   // N*48
    float*        g2     = g1 + (long)NNODES * HID;         // N*32

    // zero: acc1 + acc2 + degcnt are contiguous at the front of ws
    size_t zero_bytes = (size_t)NNODES * (HID + NACT + 1) * sizeof(float);
    hipMemsetAsync(d_ws, 0, zero_bytes, stream);
    hipMemsetAsync(d_out, 0, NACT * sizeof(float), stream);

    // degrees + normalization
    deg_kernel<<<(NEDGES + 255) / 256, 256, 0, stream>>>(col, degcnt, NEDGES);
    dinv_kernel<<<(NNODES + 255) / 256, 256, 0, stream>>>(degcnt, dinv, NNODES);

    // conv1 dense part: g1 = dinv * (x^T @ W1); x is [K][N] -> SK=NNODES, SN=1
    gemm_wmma<NOBS, HID, (long)NNODES, 1L>
        <<<NNODES / 16, 32 * (HID / 16), 0, stream>>>(x, W1, dinv, g1);

    // conv1 message passing
    {
        long total = (long)NEDGES * HID;
        scatter_kernel<HID><<<(unsigned)((total + 255) / 256), 256, 0, stream>>>(g1, row, col, acc1, total);
    }
    {
        long total = (long)NNODES * HID;
        post1_kernel<<<(unsigned)((total + 255) / 256), 256, 0, stream>>>(acc1, g1, dinv, b1, total);
    }

    // conv2 dense part: g2 = dinv * (h1 @ W2); h1 (=acc1) is row-major [N][HID] -> SK=1, SN=HID
    gemm_wmma<HID, NACT, 1L, (long)HID>
        <<<NNODES / 16, 32 * (NACT / 16), 0, stream>>>(acc1, W2, dinv, g2);

    // conv2 message passing
    {
        long total = (long)NEDGES * NACT;
        scatter_kernel<NACT><<<(unsigned)((total + 255) / 256), 256, 0, stream>>>(g2, row, col, acc2, total);
    }

    // final weighted reduce (folds conv2 self-loop + bias)
    final_reduce<<<256, 256, 0, stream>>>(g2, acc2, dinv, b2, onehot, out, NNODES);
}